// BoundaryDistanceLoss_5660766896138
// MI455X (gfx1250) — compile-verified
//
#include <hip/hip_runtime.h>
#include <math.h>

#define HH 1024
#define WW 1024
#define HWSZ (HH * WW)

typedef float v2f __attribute__((ext_vector_type(2)));
typedef float v8f __attribute__((ext_vector_type(8)));
typedef unsigned int v4u __attribute__((ext_vector_type(4)));
typedef int v4i __attribute__((ext_vector_type(4)));
typedef int v8i __attribute__((ext_vector_type(8)));

// ---------------------------------------------------------------------------
// Kernel 1: edges = seg - erode3x3(seg)   (binary input, zero padding)
// ---------------------------------------------------------------------------
__global__ void edges_kernel(const float* __restrict__ seg,
                             float* __restrict__ edges) {
    int idx = blockIdx.x * 256 + threadIdx.x;
    if (idx >= HWSZ) return;
    int r = idx >> 10;
    int c = idx & (WW - 1);
    float s = seg[idx];
    float er = 0.0f;
    if (r > 0 && r < HH - 1 && c > 0 && c < WW - 1) {
        float sum9 = 0.0f;
#pragma unroll
        for (int dr = -1; dr <= 1; ++dr)
#pragma unroll
            for (int dc = -1; dc <= 1; ++dc)
                sum9 += seg[(r + dr) * WW + (c + dc)];
        er = (sum9 == 9.0f) ? 1.0f : 0.0f;
    }
    edges[idx] = s - er;
}

// ---------------------------------------------------------------------------
// Kernel 2: per-row 1D squared distance to nearest edge pixel.
// One thread per (image,row): 2 images * 1024 rows = 2048 threads.
// g2[r][j] = min(min(left,right), BIG)^2   with BIG = 1e6 (matches reference)
// ---------------------------------------------------------------------------
__global__ void rowdist_kernel(const float* __restrict__ edges_both,
                               float* __restrict__ g2_both) {
    int tid = blockIdx.x * blockDim.x + threadIdx.x; // 0..2047
    int img = tid >> 10;
    int row = tid & (HH - 1);
    const float* e = edges_both + img * HWSZ + row * WW;
    float* g = g2_both + img * HWSZ + row * WW;
    const float BIG = 1.0e6f;
    float m = -BIG;
    for (int j = 0; j < WW; ++j) {
        float cnd = (e[j] > 0.5f) ? (float)j : -BIG;
        m = fmaxf(m, cnd);
        g[j] = (float)j - m;                 // left distance (temp)
    }
    float mn = BIG;
    for (int j = WW - 1; j >= 0; --j) {
        float cnd = (e[j] > 0.5f) ? (float)j : BIG;
        mn = fminf(mn, cnd);
        float right = mn - (float)j;
        float gg = fminf(fminf(g[j], right), BIG);
        g[j] = gg * gg;                      // squared row distance
    }
}

// ---------------------------------------------------------------------------
// TDM helper: DMA one 32x128 f32 tile (row stride WW) from global into LDS.
// D# built per CDNA5 ISA ch.8:
//   group0: count=1 | lds_addr | global_addr[56:0] | type=2
//   group1: data_size=4B, tensor_dim0=tensor_dim0_stride=1024,
//           tensor_dim1=1024, tile_dim0=128, tile_dim1=32
//   groups 2/3 zero (tile_dim3=0 -> 2D tile)
// Issued by one wave only (TDM ignores EXEC); tracked by TENSORcnt.
// This toolchain exposes the 6-arg clang-23 builtin:
//   (v4u g0, v8i g1, v4i g2, v4i g3, v8i g4, i32 cpol)
// ---------------------------------------------------------------------------
__device__ __forceinline__ void tdm_load_panel(const float* gsrc,
                                               unsigned int lds_off) {
    unsigned long long ga = (unsigned long long)(uintptr_t)gsrc;
    v4u g0;
    g0[0] = 1u;                                  // count=1, user descriptor
    g0[1] = lds_off;                             // lds_addr (bytes)
    g0[2] = (unsigned int)ga;                    // global_addr[31:0]
    g0[3] = (unsigned int)((ga >> 32) & 0x01FFFFFFull) | (2u << 30); // [56:32]|type=2

    v8i g1;
    g1[0] = (int)(2u << 16);        // workgroup_mask=0, data_size=2 (4 bytes)
    g1[1] = (int)(1024u << 16);     // tensor_dim0[15:0]=1024 in bits[31:16]
    g1[2] = (int)(1024u << 16);     // tensor_dim0 hi=0; tensor_dim1[15:0]=1024
    g1[3] = (int)(128u << 16);      // tensor_dim1 hi=0; tile_dim0=128
    g1[4] = (int)32u;               // tile_dim1=32; tile_dim2=0
    g1[5] = (int)1024u;             // tensor_dim0_stride[31:0]=1024
    g1[6] = 0;                      // stride0 hi=0; tensor_dim1_stride lo=0
    g1[7] = 0;
    v4i z4 = {0, 0, 0, 0};
    v8i z8 = {0, 0, 0, 0, 0, 0, 0, 0};
    __builtin_amdgcn_tensor_load_to_lds(g0, g1, z4, z4, z8, 0);
}

// ---------------------------------------------------------------------------
// Kernel 3: column lower envelope (brute force, matches reference):
//   D[i][j] = sqrt( min_k  g2[k][j] + (i-k)^2 )
// Tile: 32 output rows x 128 columns per block (256 threads, 16 acc/thread).
// k-panel (32 rows x 128 cols of g2) DMA'd into LDS by the Tensor Data
// Mover; fma+min inner loop on VALU.
// ---------------------------------------------------------------------------
#define TI 32
#define TJ 128
#define KC 32

__global__ void envelope_kernel(const float* __restrict__ g2_both,
                                float* __restrict__ D_both) {
    int img = blockIdx.z;
    const float* g2 = g2_both + img * HWSZ;
    float* D = D_both + img * HWSZ;
    int j0 = blockIdx.x * TJ;
    int i0 = blockIdx.y * TI;
    int t = threadIdx.x;
    int jloc = t & (TJ - 1);
    int ig = t >> 7; // 0..1, selects 16-row group
    float ibase = (float)(i0 + ig * 16);

    __shared__ float panel[KC][TJ];
    unsigned int lds_off = (unsigned int)(uintptr_t)(&panel[0][0]);

    float acc[16];
#pragma unroll
    for (int m = 0; m < 16; ++m) acc[m] = 3.0e38f;

    for (int kb = 0; kb < HH; kb += KC) {
        // prefetch the next panel into caches (global_prefetch_b8):
        if (kb + KC < HH) {
            const float* nxt =
                &g2[(kb + KC + (t >> 3)) * WW + j0 + (t & 7) * 16];
            __builtin_prefetch(nxt, 0, 3);
        }
        // wave 0 owns the TDM: one tensor_load_to_lds for the whole panel
        if (t < 32) {
            tdm_load_panel(&g2[kb * WW + j0], lds_off);
            __builtin_amdgcn_s_wait_tensorcnt(0);
        }
        __syncthreads();

#pragma unroll 4
        for (int kk = 0; kk < KC; ++kk) {
            float gv = panel[kk][jloc];
            float kf = (float)(kb + kk);
#pragma unroll
            for (int m = 0; m < 16; ++m) {
                float d = (ibase + (float)m) - kf;
                acc[m] = fminf(acc[m], __builtin_fmaf(d, d, gv));
            }
        }
        __syncthreads();
    }

#pragma unroll
    for (int m = 0; m < 16; ++m) {
        D[(i0 + ig * 16 + m) * WW + j0 + jloc] = sqrtf(acc[m]);
    }
}

// ---------------------------------------------------------------------------
// Kernel 4: WMMA-based reduction  partial = sum(edges * dt) over a stripe.
// With B = ones(4x16), D=A*B+C accumulates row-sums of A; the sum of all 256
// entries of C equals 16 * sum(A). FP32 WMMA -> exact accumulation.
// Two independent accumulator chains (c0/c1) hide the WMMA->WMMA RAW hazard;
// each iteration folds 128 products via two v_wmma_f32_16x16x4_f32.
// grid = (64 blocks, 2 terms), 256 threads (8 waves), 512 waves/term total.
// ---------------------------------------------------------------------------
__global__ void wmma_dot_kernel(const float* __restrict__ e0,
                                const float* __restrict__ d0,
                                const float* __restrict__ e1,
                                const float* __restrict__ d1,
                                float* __restrict__ partials) {
    const float* e = blockIdx.y ? e1 : e0;
    const float* dv = blockIdx.y ? d1 : d0;
    int lane = threadIdx.x & 31;
    int wave = threadIdx.x >> 5;
    int gw = blockIdx.x * 8 + wave; // 0..511

    v8f c0 = {};
    v8f c1 = {};
    v2f b = {1.0f, 1.0f};

    // HWSZ/128 = 8192 double-chunks; 512 waves -> exactly 16 iters per wave
    for (int ch = gw; ch < HWSZ / 128; ch += 512) {
        int base = ch * 128;
        v2f a0, a1;
        a0[0] = e[base + lane] * dv[base + lane];
        a0[1] = e[base + 32 + lane] * dv[base + 32 + lane];
        a1[0] = e[base + 64 + lane] * dv[base + 64 + lane];
        a1[1] = e[base + 96 + lane] * dv[base + 96 + lane];
        // 8 args: (neg_a, A, neg_b, B, c_mod, C, reuse_a, reuse_b)
        c0 = __builtin_amdgcn_wmma_f32_16x16x4_f32(
            false, a0, false, b, (short)0, c0, false, false);
        c1 = __builtin_amdgcn_wmma_f32_16x16x4_f32(
            false, a1, false, b, (short)0, c1, false, false);
    }

    float s = (c0[0] + c1[0]) + (c0[1] + c1[1]) + (c0[2] + c1[2]) +
              (c0[3] + c1[3]) + (c0[4] + c1[4]) + (c0[5] + c1[5]) +
              (c0[6] + c1[6]) + (c0[7] + c1[7]);
    __shared__ float red[256];
    red[threadIdx.x] = s;
    __syncthreads();
    for (int off = 128; off > 0; off >>= 1) {
        if (threadIdx.x < (unsigned)off) red[threadIdx.x] += red[threadIdx.x + off];
        __syncthreads();
    }
    if (threadIdx.x == 0)
        partials[blockIdx.y * 64 + blockIdx.x] = red[0] * (1.0f / 16.0f);
}

// ---------------------------------------------------------------------------
// Kernel 5: deterministic final reduction + sigmoid
// partials[0..63] = term0 blocks, partials[64..127] = term1 blocks
// ---------------------------------------------------------------------------
__global__ void finalize_kernel(const float* __restrict__ partials,
                                float* __restrict__ out) {
    __shared__ float red[128];
    int t = threadIdx.x; // 128 threads
    red[t] = partials[t];
    __syncthreads();
    for (int off = 32; off > 0; off >>= 1) {
        if ((t & 63) < off) red[t] += red[t + off];
        __syncthreads();
    }
    if (t == 0) {
        float s0 = red[0];
        float s1 = red[64];
        float invN = 1.0f / (float)HWSZ;
        float loss = 0.5f * (s0 * invN + s1 * invN);
        out[0] = 1.0f / (1.0f + expf(-loss));
    }
}

// ---------------------------------------------------------------------------
// Launch
// ---------------------------------------------------------------------------
extern "C" void kernel_launch(void* const* d_in, const int* in_sizes, int n_in,
                              void* d_out, int out_size, void* d_ws, size_t ws_size,
                              hipStream_t stream) {
    const float* preds = (const float*)d_in[0];
    const float* targets = (const float*)d_in[1];
    float* out = (float*)d_out;

    float* ws = (float*)d_ws;
    float* edges0 = ws;                // HWSZ floats (pred edges)
    float* edges1 = ws + HWSZ;         // HWSZ floats (target edges)
    float* g2 = ws + 2 * (size_t)HWSZ; // 2*HWSZ floats (both images)
    float* D = ws + 4 * (size_t)HWSZ;  // 2*HWSZ floats (both dt maps)
    float* partials = ws + 6 * (size_t)HWSZ; // 128 floats

    edges_kernel<<<HWSZ / 256, 256, 0, stream>>>(preds, edges0);
    edges_kernel<<<HWSZ / 256, 256, 0, stream>>>(targets, edges1);

    rowdist_kernel<<<8, 256, 0, stream>>>(edges0, g2);

    dim3 eg(WW / TJ, HH / TI, 2);
    envelope_kernel<<<eg, 256, 0, stream>>>(g2, D);

    // pred_err   = target_edges * pred_dt   -> (edges1, D[img0])
    // target_err = pred_edges  * target_dt  -> (edges0, D[img1])
    dim3 dg(64, 2);
    wmma_dot_kernel<<<dg, 256, 0, stream>>>(edges1, D, edges0, D + HWSZ,
                                            partials);

    finalize_kernel<<<1, 128, 0, stream>>>(partials, out);
}